// GIN_82841329205346
// MI455X (gfx1250) — compile-verified
//
#include <hip/hip_runtime.h>
#include <hip/hip_bf16.h>

#define N_NODES   50000
#define N_EDGES   800000
#define NFEAT     128
#define DIM       96
#define NGRAPHS   512
#define CATDIM    (DIM * 3)
#define BN_EPS    1e-5f

typedef __attribute__((ext_vector_type(2))) float v2f;
typedef __attribute__((ext_vector_type(8))) float v8f;

// ---------------- utility ----------------
__global__ void zero_kernel(float* __restrict__ p, int n) {
    int i = blockIdx.x * blockDim.x + threadIdx.x;
    if (i < n) p[i] = 0.0f;
}

// z[n, d] = h[n, d]   (h has row stride hstride, z has row stride din)
__global__ void copy_kernel(float* __restrict__ z, const float* __restrict__ h,
                            int hstride, int din, int n_nodes) {
    int i = blockIdx.x * blockDim.x + threadIdx.x;
    if (i >= n_nodes * din) return;
    int n = i / din, d = i - n * din;
    z[(size_t)n * din + d] = h[(size_t)n * hstride + d];
}

// z[dst[e], c..c+3] += h[src[e], c..c+3] * w[e]
__global__ void scatter_kernel(const float* __restrict__ h, int hstride, int din,
                               const int* __restrict__ src, const int* __restrict__ dst,
                               const float* __restrict__ w, float* __restrict__ z,
                               int n_edges) {
    int chunks = din >> 2;
    int idx = blockIdx.x * blockDim.x + threadIdx.x;
    if (idx >= n_edges * chunks) return;
    int e = idx / chunks;
    int c = (idx - e * chunks) << 2;
    int s = src[e], t = dst[e];
    float we = w[e];
    const float4 hv = *(const float4*)(h + (size_t)s * hstride + c);
    float* zp = z + (size_t)t * din + c;
    atomicAdd(zp + 0, hv.x * we);
    atomicAdd(zp + 1, hv.y * we);
    atomicAdd(zp + 2, hv.z * we);
    atomicAdd(zp + 3, hv.w * we);
}

// ---------------- WMMA GEMM: out = relu(A[N,K] @ W[K,96] + bias) ----------------
// 256 threads = 8 waves; each wave computes a 16-row x 96-col strip with six
// V_WMMA_F32_16X16X4_F32 accumulators. W is staged TRANSPOSED in LDS
// (sWt[col][k], row stride K+4) so each B operand is one aligned ds_load_b64
// with a conflict-free bank pattern. Optionally accumulates per-feature
// sum / sum-of-squares (for fused BatchNorm statistics).
__global__ __launch_bounds__(256) void gemm_relu_kernel(
    const float* __restrict__ A, int K,
    const float* __restrict__ W, const float* __restrict__ bias,
    float* __restrict__ out, float* __restrict__ stats,  // stats: [sum(96), sq(96)] or null
    int n_rows, int doStats) {
    __shared__ float sWt[DIM * (NFEAT + 4)];   // 96 x 132 floats = 50.7 KB max
    __shared__ float sSum[DIM];
    __shared__ float sSq[DIM];

    const int KP = K + 4;                      // padded LDS row stride (bank decorrelation)
    int tid = threadIdx.x;
    for (int i = tid; i < DIM * K; i += blockDim.x) {
        int col = i / K, k = i - col * K;
        sWt[col * KP + k] = W[k * DIM + col];
    }
    if (tid < DIM) { sSum[tid] = 0.0f; sSq[tid] = 0.0f; }
    __syncthreads();

    int wave = tid >> 5;
    int lane = tid & 31;
    int m  = lane & 15;     // row within tile (A) / column within tile (B,C,D)
    int kh = lane >> 4;     // K-half select: lanes 16-31 hold K+2,K+3
    int row0 = (blockIdx.x * 8 + wave) * 16;
    int row  = row0 + m;
    if (row >= n_rows) row = n_rows - 1;   // clamp loads; stores masked (EXEC stays full)
    const float* arow = A + (size_t)row * K;

    v8f acc[6];
#pragma unroll
    for (int j = 0; j < 6; ++j) acc[j] = 0;

    for (int k0 = 0; k0 < K; k0 += 4) {
        int ka = k0 + 2 * kh;                       // always even -> 8B aligned
        v2f a = *(const v2f*)(arow + ka);           // global_load_b64
#pragma unroll
        for (int j = 0; j < 6; ++j) {
            v2f b = *(const v2f*)(&sWt[(j * 16 + m) * KP + ka]);  // ds_load_b64
            acc[j] = __builtin_amdgcn_wmma_f32_16x16x4_f32(
                false, a, false, b, (short)0, acc[j], false, false);
        }
    }

    // C/D layout: VGPR r -> M = r (lanes 0-15), M = r+8 (lanes 16-31); N = lane&15
#pragma unroll
    for (int j = 0; j < 6; ++j) {
        int col = j * 16 + m;
        float bv = bias[col];
        float lsum = 0.0f, lsq = 0.0f;
#pragma unroll
        for (int r = 0; r < 8; ++r) {
            int mm = row0 + r + 8 * kh;
            float v = fmaxf(acc[j][r] + bv, 0.0f);
            if (mm < n_rows) {
                out[(size_t)mm * DIM + col] = v;
                lsum += v;
                lsq  += v * v;
            }
        }
        if (doStats) {
            atomicAdd(&sSum[col], lsum);
            atomicAdd(&sSq[col],  lsq);
        }
    }
    __syncthreads();
    if (doStats) {
        for (int i = tid; i < DIM; i += blockDim.x) {
            atomicAdd(&stats[i],        sSum[i]);
            atomicAdd(&stats[DIM + i],  sSq[i]);
        }
    }
}

// ---------------- BatchNorm apply + concat write + global_add_pool ----------------
__global__ void bn_pool_kernel(const float* __restrict__ z, const float* __restrict__ stats,
                               const float* __restrict__ gamma, const float* __restrict__ beta,
                               const int* __restrict__ batch,
                               float* __restrict__ x_nodes, float* __restrict__ x_g,
                               int layer, int n_nodes) {
    int i = blockIdx.x * blockDim.x + threadIdx.x;
    if (i >= n_nodes * DIM) return;
    int n = i / DIM, d = i - n * DIM;
    float invN = 1.0f / (float)n_nodes;
    float mean = stats[d] * invN;
    float var  = stats[DIM + d] * invN - mean * mean;
    float inv  = rsqrtf(var + BN_EPS);
    float v = (z[i] - mean) * inv * gamma[d] + beta[d];
    int colbase = layer * DIM + d;
    x_nodes[(size_t)n * CATDIM + colbase] = v;
    atomicAdd(&x_g[(size_t)batch[n] * CATDIM + colbase], v);
}

// ---------------- driver ----------------
extern "C" void kernel_launch(void* const* d_in, const int* in_sizes, int n_in,
                              void* d_out, int out_size, void* d_ws, size_t ws_size,
                              hipStream_t stream) {
    const float* x     = (const float*)d_in[0];
    const float* ew    = (const float*)d_in[1];
    const int*   ei    = (const int*)d_in[2];   // [2, E]
    const int*   batch = (const int*)d_in[3];
    const int* src = ei;
    const int* dst = ei + N_EDGES;

    float* x_g     = (float*)d_out;                       // [512, 288]
    float* x_nodes = (float*)d_out + NGRAPHS * CATDIM;    // [50000, 288]

    float* z     = (float*)d_ws;                          // [N, 128] worst case
    float* t1    = z  + (size_t)N_NODES * NFEAT;          // [N, 96]
    float* t2    = t1 + (size_t)N_NODES * DIM;            // [N, 96]
    float* stats = t2 + (size_t)N_NODES * DIM;            // [192]

    // zero pooled output
    {
        int n = NGRAPHS * CATDIM;
        zero_kernel<<<(n + 255) / 256, 256, 0, stream>>>(x_g, n);
    }

    for (int L = 0; L < 3; ++L) {
        const float* W1    = (const float*)d_in[4 + 6 * L + 0];
        const float* b1    = (const float*)d_in[4 + 6 * L + 1];
        const float* W2    = (const float*)d_in[4 + 6 * L + 2];
        const float* b2    = (const float*)d_in[4 + 6 * L + 3];
        const float* gamma = (const float*)d_in[4 + 6 * L + 4];
        const float* beta  = (const float*)d_in[4 + 6 * L + 5];

        int din = (L == 0) ? NFEAT : DIM;
        const float* h = (L == 0) ? x : (x_nodes + (L - 1) * DIM);
        int hstride    = (L == 0) ? NFEAT : CATDIM;

        // z = h  (the "(1+eps)*h" term, eps = 0)
        {
            int n = N_NODES * din;
            copy_kernel<<<(n + 255) / 256, 256, 0, stream>>>(z, h, hstride, din, N_NODES);
        }
        // z += segment_sum( h[src] * w )
        {
            int n = N_EDGES * (din >> 2);
            scatter_kernel<<<(n + 255) / 256, 256, 0, stream>>>(h, hstride, din,
                                                                src, dst, ew, z, N_EDGES);
        }
        // t1 = relu(z @ W1 + b1)
        {
            int grid = (N_NODES + 127) / 128;
            gemm_relu_kernel<<<grid, 256, 0, stream>>>(z, din, W1, b1, t1, nullptr,
                                                       N_NODES, 0);
        }
        // stats = 0 ; t2 = relu(t1 @ W2 + b2) with fused sum / sumsq
        zero_kernel<<<1, 256, 0, stream>>>(stats, 2 * DIM);
        {
            int grid = (N_NODES + 127) / 128;
            gemm_relu_kernel<<<grid, 256, 0, stream>>>(t1, DIM, W2, b2, t2, stats,
                                                       N_NODES, 1);
        }
        // BatchNorm + write concat slice + pool
        {
            int n = N_NODES * DIM;
            bn_pool_kernel<<<(n + 255) / 256, 256, 0, stream>>>(t2, stats, gamma, beta,
                                                                batch, x_nodes, x_g,
                                                                L, N_NODES);
        }
    }
}